// Get_ltpe_11751030522907
// MI455X (gfx1250) — compile-verified
//
#include <hip/hip_runtime.h>
#include <stdint.h>

#define TILE_W 64
#define TILE_H 16
#define HALO_H 18
#define ROWS   72                 // padded LDS row stride (floats)
#define PLANE  (HALO_H * ROWS)    // 1296 floats per channel plane
#define IMG    1048576            // 1024*1024

__device__ __forceinline__ void async_ld_b128(uint32_t lds_byte_off, const float* gptr) {
    // VDST = LDS byte offset (per-lane), VADDR = 64-bit global address, no SADDR.
    asm volatile("global_load_async_to_lds_b128 %0, %1, off"
                 :: "v"(lds_byte_off), "v"(gptr)
                 : "memory");
}

__device__ __forceinline__ void wait_asynccnt0() {
    asm volatile("s_wait_asynccnt 0" ::: "memory");
}

// Store with TH=WB: keep line dirty in far cache (L2), override write-rinse.
// The 32MB prenorm plane then lives purely in the 192MB L2 between kernels.
__device__ __forceinline__ void store_b32_wb(float* gptr, float v) {
    asm volatile("global_store_b32 %0, %1, off th:TH_STORE_WB"
                 :: "v"(gptr), "v"(v)
                 : "memory");
}

__global__ void ltpe_init(double* __restrict__ acc) {
    if (threadIdx.x < 16) acc[threadIdx.x] = 0.0;
}

__global__ __launch_bounds__(256) void ltpe_main(const float* __restrict__ x,
                                                 float* __restrict__ out,
                                                 double* __restrict__ acc) {
    __shared__ __align__(16) float lds_x[3 * PLANE];
    __shared__ __align__(16) float lds_g[PLANE];
    __shared__ float red_s[256];
    __shared__ float red_q[256];

    const int tid = threadIdx.x;
    const int b   = blockIdx.z;
    const int c0  = blockIdx.x * TILE_W;
    const int r0  = blockIdx.y * TILE_H;
    const float* xb = x + (size_t)b * 3 * IMG;

    // ---- interior tile loads: 3 ch * 18 rows * 16 float4 = 864, async -> LDS ----
    for (int i = tid; i < 864; i += 256) {
        int ch  = i / 288;
        int rem = i - ch * 288;
        int row = rem >> 4;
        int q   = rem & 15;
        int gr  = r0 - 1 + row;
        float* lptr = &lds_x[ch * PLANE + row * ROWS + 4 + q * 4];
        if (gr >= 0 && gr < 1024) {
            const float* gp = xb + (size_t)ch * IMG + (size_t)gr * 1024 + c0 + q * 4;
            async_ld_b128((uint32_t)(uintptr_t)lptr, gp);
        } else {
            float4 z = make_float4(0.f, 0.f, 0.f, 0.f);
            *(float4*)lptr = z;   // zero padding rows outside image
        }
    }
    // ---- halo columns (gc = c0-1 and c0+64): 3 ch * 18 rows * 2 = 108 ----
    if (tid < 108) {
        int side = tid & 1;
        int idx  = tid >> 1;          // 0..53
        int ch   = idx / 18;
        int row  = idx - ch * 18;
        int gr   = r0 - 1 + row;
        int gc   = side ? (c0 + TILE_W) : (c0 - 1);
        float v  = 0.f;
        if (gr >= 0 && gr < 1024 && gc >= 0 && gc < 1024)
            v = xb[(size_t)ch * IMG + (size_t)gr * 1024 + gc];
        lds_x[ch * PLANE + row * ROWS + (side ? (4 + TILE_W) : 3)] = v;
    }

    wait_asynccnt0();        // my wave's async LDS writes done
    __syncthreads();         // everyone's done

    // ---- grayscale into LDS: 18 rows * 66 cols = 1188 ----
    for (int i = tid; i < 1188; i += 256) {
        int row = i / 66;
        int col = 3 + (i - row * 66);
        int k   = row * ROWS + col;
        lds_g[k] = 0.30f * lds_x[k] + 0.59f * lds_x[PLANE + k] + 0.11f * lds_x[2 * PLANE + k];
    }
    __syncthreads();

    // ---- 8-neighbor LTP stencil; write prenorm into channel-0 plane of out ----
    float s = 0.f, sq = 0.f;
    float* outp = out + (size_t)(b * 3) * IMG;   // channel 0
    #pragma unroll
    for (int k = 0; k < 4; ++k) {
        int p    = tid + k * 256;
        int prow = p >> 6;
        int pcol = p & 63;
        int base = (1 + prow) * ROWS + (4 + pcol);
        float g  = lds_g[base];
        float a;
        a  = (  1.0f/255.0f) * (g - lds_g[base - 1]        + 1.0f);  // ( 0,-1)
        a += (  2.0f/255.0f) * (g - lds_g[base + ROWS - 1] + 1.0f);  // ( 1,-1)
        a += (  4.0f/255.0f) * (g - lds_g[base + ROWS]     + 1.0f);  // ( 1, 0)
        a += (  8.0f/255.0f) * (g - lds_g[base + ROWS + 1] + 1.0f);  // ( 1, 1)
        a += ( 16.0f/255.0f) * (g - lds_g[base + 1]        + 1.0f);  // ( 0, 1)
        a += ( 32.0f/255.0f) * (g - lds_g[base - ROWS + 1] + 1.0f);  // (-1, 1)
        a += ( 64.0f/255.0f) * (g - lds_g[base - ROWS]     + 1.0f);  // (-1, 0)
        a += (128.0f/255.0f) * (g - lds_g[base - ROWS - 1] + 1.0f);  // (-1,-1)
        a *= 0.5f;
        store_b32_wb(&outp[(size_t)(r0 + prow) * 1024 + c0 + pcol], a); // stay dirty in L2
        s  += a;
        sq += a * a;
    }

    // ---- block reduction of sum / sumsq, then f64 atomics per image ----
    red_s[tid] = s;
    red_q[tid] = sq;
    __syncthreads();
    for (int off = 128; off > 0; off >>= 1) {
        if (tid < off) {
            red_s[tid] += red_s[tid + off];
            red_q[tid] += red_q[tid + off];
        }
        __syncthreads();
    }
    if (tid == 0) {
        atomicAdd(&acc[b],     (double)red_s[0]);
        atomicAdd(&acc[8 + b], (double)red_q[0]);
    }
}

__global__ void ltpe_stats(const double* __restrict__ acc, float* __restrict__ st) {
    int t = threadIdx.x;
    if (t < 8) {
        const double N = 1048576.0;
        double mean = acc[t] / N;
        double var  = acc[8 + t] / N - mean * mean;
        st[t]     = (float)mean;
        st[8 + t] = (float)rsqrt(var + 1e-5);
    }
}

__global__ __launch_bounds__(256) void ltpe_norm(float* __restrict__ out,
                                                 const float* __restrict__ st) {
    // 8192 blocks: 1024 blocks per image plane -> b is uniform per block (scalar loads)
    int    b  = blockIdx.x >> 10;
    size_t pi = (size_t)(blockIdx.x & 1023) * 256 + threadIdx.x;  // float4 idx in plane
    float4* base = (float4*)(out + (size_t)b * 3 * IMG);
    float  m   = st[b];
    float  inv = st[8 + b];
    float4 v = base[pi];                 // hits L2 (prenorm kept dirty there)
    v.x = (v.x - m) * inv;
    v.y = (v.y - m) * inv;
    v.z = (v.z - m) * inv;
    v.w = (v.w - m) * inv;
    base[pi]          = v;   // ch0 (overwrites prenorm line in place)
    base[pi + 262144] = v;   // ch1
    base[pi + 524288] = v;   // ch2
}

extern "C" void kernel_launch(void* const* d_in, const int* in_sizes, int n_in,
                              void* d_out, int out_size, void* d_ws, size_t ws_size,
                              hipStream_t stream) {
    (void)in_sizes; (void)n_in; (void)out_size; (void)ws_size;
    const float* x   = (const float*)d_in[0];
    float*       out = (float*)d_out;
    double*      acc = (double*)d_ws;          // 16 doubles: sum[8], sumsq[8]
    float*       st  = (float*)(acc + 16);     // 16 floats:  mean[8], inv[8]

    ltpe_init<<<1, 32, 0, stream>>>(acc);
    dim3 grid(1024 / TILE_W, 1024 / TILE_H, 8);
    ltpe_main<<<grid, 256, 0, stream>>>(x, out, acc);
    ltpe_stats<<<1, 32, 0, stream>>>(acc, st);
    ltpe_norm<<<8192, 256, 0, stream>>>(out, st);
}